// GeneralOdeLayer_32873679684211
// MI455X (gfx1250) — compile-verified
//
#include <hip/hip_runtime.h>
#include <cstdint>
#include <cstddef>

// ---------------------------------------------------------------------------
// GeneralOdeLayer: S=65536 independent linear 2-state ODEs, RK4, T=128
// intervals x 16 substeps. For x' = A x (A lower-triangular 2x2, constant
// per system), RK4 with substep h is EXACTLY x <- R(h*A) x, with R the
// degree-4 Taylor polynomial of exp. 16 substeps == R(hA)^16 == 4 squarings
// of a lower-triangular 2x2. This makes the kernel memory-bound (~98 MB at
// 23.3 TB/s), so the implementation is organized around data movement:
//   - TDM (tensor_load_to_lds) streams t_vals tiles [256 x 8] into LDS with
//     hardware row padding (stride 9 floats -> conflict-free), double-buffered
//   - output staged in LDS, stored with coalesced float4 rows
// ---------------------------------------------------------------------------

typedef uint32_t v4u __attribute__((ext_vector_type(4)));
typedef int      v8i __attribute__((ext_vector_type(8)));
typedef int      v4i __attribute__((ext_vector_type(4)));

#define S_TOTAL 65536
#define T_TOTAL 128
#define BLK     256
#define TC      8                 // intervals per chunk
#define NCH     (T_TOTAL / TC)    // 16 chunks
#define TROW    (TC + 1)          // t-tile LDS row stride (TDM pad: 8 dw + 1 dw)
#define OROW    20                // out-tile LDS row stride (16 data + 4 pad, 16B aligned)

// Issue one TDM 2D tile load: t_vals[s0 .. s0+255][chunk*TC .. chunk*TC+TC)
// into LDS at lds_dst, with hardware padding of 1 DWORD every 8 DWORDs
// (-> row stride TROW floats). Descriptor per CDNA5 ISA §8 (D# groups 0/1).
__device__ __forceinline__ void tdm_load_ttile(const float* t_vals, int s0, int chunk,
                                               float* lds_dst)
{
    uint64_t gaddr = (uint64_t)(uintptr_t)t_vals +
                     ((uint64_t)s0 * T_TOTAL + (uint64_t)chunk * TC) * sizeof(float);
    uint32_t lds_addr = (uint32_t)(uintptr_t)lds_dst;   // low 32 bits = LDS byte addr

    v4u g0;
    g0.x = 1u;                                           // count=1, user descriptor
    g0.y = lds_addr;                                     // lds_addr
    g0.z = (uint32_t)gaddr;                              // global_addr[31:0]
    g0.w = ((uint32_t)(gaddr >> 32) & 0x01FFFFFFu)       // global_addr[56:32]
           | (2u << 30);                                 // type=2 ("image")

    v8i g1;
    g1.s0 = (int)((2u << 16)      // data_size = 4 bytes
                | (1u << 20)      // pad_enable
                | (2u << 22)      // pad_interval: 8 DWORDs
                | (0u << 25));    // pad_amount: 1 DWORD
    g1.s1 = (int)(((uint32_t)T_TOTAL & 0xFFFFu) << 16);  // tensor_dim0 lo16 = 128
    g1.s2 = 0;                                           // dim0 hi16=0, tensor_dim1 lo16=0
    g1.s3 = (int)(((uint32_t)S_TOTAL >> 16)              // tensor_dim1 hi16 = 1
                | ((uint32_t)TC << 16));                 // tile_dim0 = 8
    g1.s4 = (int)BLK;                                    // tile_dim1 = 256, tile_dim2 = 0
    g1.s5 = (int)T_TOTAL;                                // tensor_dim0_stride = 128
    g1.s6 = 0;
    g1.s7 = 0;

    v4i g2 = {0, 0, 0, 0};        // 2D tensor: groups 2/3 unused (tile_dim2..4 = 0)
    v4i g3 = {0, 0, 0, 0};
    v8i g4 = {0, 0, 0, 0, 0, 0, 0, 0};   // extra group required by 6-arg builtin form

    __builtin_amdgcn_tensor_load_to_lds(g0, g1, g2, g3, g4, 0);
}

__global__ __launch_bounds__(BLK)
void ode_linrk4_kernel(const float* __restrict__ k1,
                       const float* __restrict__ k2,
                       const float* __restrict__ x0s,
                       const float* __restrict__ t_vals,
                       float* __restrict__ out)
{
    __shared__ float t_tile[2][BLK * TROW];
    __shared__ float out_tile[BLK * OROW];

    const int tid = threadIdx.x;
    const int s0  = blockIdx.x * BLK;
    const int s   = s0 + tid;

    const float k1v = k1[s];
    const float k2v = k2[s];
    float x0 = x0s[2 * s + 0];
    float x1 = x0s[2 * s + 1];
    float t_prev = 0.0f;

    const bool issuer = (tid < 32);   // wave 0 drives the TDM

    if (issuer) {
        tdm_load_ttile(t_vals, s0, 0, &t_tile[0][0]);
        __builtin_amdgcn_s_wait_tensorcnt(0);
    }
    __syncthreads();

    const float inv6  = 1.0f / 6.0f;
    const float inv24 = 1.0f / 24.0f;

    for (int c = 0; c < NCH; ++c) {
        const int buf = c & 1;

        // Prefetch next chunk asynchronously (TDM), overlapping compute.
        if (issuer && (c + 1) < NCH) {
            tdm_load_ttile(t_vals, s0, c + 1, &t_tile[buf ^ 1][0]);
        }

        const float* trow = &t_tile[buf][tid * TROW];
        float*       orow = &out_tile[tid * OROW];

        #pragma unroll
        for (int t = 0; t < TC; ++t) {
            const float t1 = trow[t];
            const float h  = (t1 - t_prev) * 0.0625f;   // substep = (t1-t0)/16
            t_prev = t1;

            const float a = -h * k1v;
            const float d = -h * k2v;
            // Degree-4 Taylor of exp == RK4 amplification polynomial.
            const float pa = 1.0f + a * (1.0f + a * (0.5f + a * (inv6 + a * inv24)));
            const float pd = 1.0f + d * (1.0f + d * (0.5f + d * (inv6 + d * inv24)));
            const float e1 = a + d;
            const float aa = a * a, dd = d * d, ad = a * d;
            // Off-diag polynomial (exact, no division):
            // q = 1 + (a+d)/2 + (a^2+ad+d^2)/6 + (a+d)(a^2+d^2)/24
            const float q = 1.0f + 0.5f * e1 + inv6 * (aa + ad + dd)
                          + inv24 * (e1 * (aa + dd));

            float A = pa, D = pd, C = (-a) * q;         // C entry: c = h*k1 = -a
            #pragma unroll
            for (int k = 0; k < 4; ++k) {               // P^16 (16 substeps)
                C = C * (A + D);
                A = A * A;
                D = D * D;
            }
            const float nx0 = A * x0;
            const float nx1 = C * x0 + D * x1;
            x0 = nx0;
            x1 = nx1;
            orow[2 * t]     = x0;
            orow[2 * t + 1] = x1;
        }
        __syncthreads();   // out_tile complete (all waves)

        // Coalesced store of the [256 x 16]f tile as float4 rows of 4.
        {
            float4* og = (float4*)(out + (size_t)s0 * (T_TOTAL * 2) + (size_t)c * (TC * 2));
            #pragma unroll
            for (int k = 0; k < 4; ++k) {
                const int j    = tid + k * BLK;   // 0..1023 float4 elements
                const int row  = j >> 2;
                const int col4 = j & 3;
                const float4 v = *(const float4*)&out_tile[row * OROW + col4 * 4];
                og[(size_t)row * (T_TOTAL * 2 / 4) + col4] = v;
            }
        }

        // Ensure next t-tile landed in LDS before anyone reads it, and that
        // all out_tile reads are done before the next chunk overwrites it.
        if (issuer && (c + 1) < NCH) {
            __builtin_amdgcn_s_wait_tensorcnt(0);
        }
        __syncthreads();
    }
}

extern "C" void kernel_launch(void* const* d_in, const int* in_sizes, int n_in,
                              void* d_out, int out_size, void* d_ws, size_t ws_size,
                              hipStream_t stream) {
    const float* k1     = (const float*)d_in[0];
    const float* k2     = (const float*)d_in[1];
    const float* x0s    = (const float*)d_in[2];
    const float* t_vals = (const float*)d_in[3];
    float*       out    = (float*)d_out;

    dim3 grid(S_TOTAL / BLK);
    dim3 block(BLK);
    ode_linrk4_kernel<<<grid, block, 0, stream>>>(k1, k2, x0s, t_vals, out);
}